// GATSign_70995809403007
// MI455X (gfx1250) — compile-verified
//
#include <hip/hip_runtime.h>
#include <stdint.h>

#define DIM 64
#define WAVES_PER_BLOCK 8
#define ROWS_PER_BLOCK (WAVES_PER_BLOCK * 16)
#define NEG_SLOPE 0.2f

typedef float v2f __attribute__((ext_vector_type(2)));
typedef float v8f __attribute__((ext_vector_type(8)));

// ---------------------------------------------------------------------------
// float <-> order-preserving uint mapping for atomicMax on floats
// ---------------------------------------------------------------------------
__device__ __forceinline__ unsigned mapf(float f) {
    unsigned b = __float_as_uint(f);
    return (b & 0x80000000u) ? ~b : (b | 0x80000000u);
}
__device__ __forceinline__ float unmapf(unsigned u) {
    unsigned b = (u & 0x80000000u) ? (u & 0x7FFFFFFFu) : ~u;
    return __uint_as_float(b);
}

__device__ __forceinline__ void get_edge(long long e, const long long* __restrict__ pos,
                                         const long long* __restrict__ neg, long long ne,
                                         int& s, int& d) {
    if (e < ne) {                 // pos edges: row0 = src, row1 = dst
        s = (int)pos[e];
        d = (int)pos[ne + e];
    } else if (e < 2 * ne) {      // neg edges
        long long k = e - ne;
        s = (int)neg[k];
        d = (int)neg[ne + k];
    } else {                      // self loops
        s = d = (int)(e - 2 * ne);
    }
}

// ---------------------------------------------------------------------------
// Kernel 1: h = z @ W via V_WMMA_F32_16X16X4_F32, fused alpha_s / alpha_d
// One wave computes a 16x64 tile of h (4 col-tiles x 16 K-steps of K=4).
// ---------------------------------------------------------------------------
__global__ __launch_bounds__(256) void gat_gemm_alpha(
    const float* __restrict__ z, const float* __restrict__ W,
    const float* __restrict__ avs, const float* __restrict__ avd,
    float* __restrict__ h, float* __restrict__ alpha_s, float* __restrict__ alpha_d,
    int n)
{
    __shared__ float sW[DIM * DIM];                     // 16 KB
    __shared__ float sH[WAVES_PER_BLOCK][16][DIM + 1];  // padded, ~33 KB

    int tid = threadIdx.x;
    // stage W into LDS (whole block)
    #pragma unroll
    for (int i = 0; i < 16; ++i) sW[tid + 256 * i] = W[tid + 256 * i];
    __syncthreads();

    int wave = tid >> 5;
    int lane = tid & 31;
    int hf   = lane >> 4;    // lane half (0: lanes 0-15, 1: lanes 16-31)
    int l16  = lane & 15;
    int rowbase = blockIdx.x * ROWS_PER_BLOCK + wave * 16;
    if (rowbase >= n) return;   // wave-uniform exit; EXEC stays all-ones for WMMA

    v8f acc[4];
    #pragma unroll
    for (int t = 0; t < 4; ++t) acc[t] = (v8f){};

#if __has_builtin(__builtin_amdgcn_wmma_f32_16x16x4_f32)
    // A fragment (16x4 f32): lanes 0-15 -> K = {4kk+0, 4kk+1}, lanes 16-31 -> K = {4kk+2, 4kk+3}
    int arow = rowbase + l16; if (arow >= n) arow = n - 1;   // clamp, keep EXEC full
    const float* zr = z + (size_t)arow * DIM;
    #pragma unroll
    for (int kk = 0; kk < 16; ++kk) {
        int k0 = 4 * kk + 2 * hf;
        v2f a;
        a.x = zr[k0];
        a.y = zr[k0 + 1];
        #pragma unroll
        for (int t = 0; t < 4; ++t) {
            // B fragment (4x16 f32), row-striped across lanes mirroring A's K split
            v2f b;
            b.x = sW[(k0    ) * DIM + 16 * t + l16];
            b.y = sW[(k0 + 1) * DIM + 16 * t + l16];
            acc[t] = __builtin_amdgcn_wmma_f32_16x16x4_f32(
                /*neg_a=*/false, a, /*neg_b=*/false, b,
                /*c_mod=*/(short)0, acc[t], /*reuse_a=*/false, /*reuse_b=*/false);
        }
    }
#else
    // scalar fallback producing the same C-fragment layout
    #pragma unroll
    for (int t = 0; t < 4; ++t) {
        #pragma unroll
        for (int r = 0; r < 8; ++r) {
            int row = rowbase + r + 8 * hf; if (row >= n) row = n - 1;
            float s = 0.f;
            for (int k = 0; k < DIM; ++k)
                s += z[(size_t)row * DIM + k] * sW[k * DIM + 16 * t + l16];
            acc[t][r] = s;
        }
    }
#endif

    // C/D layout: VGPR r holds row (r) for lanes 0-15 and row (r+8) for lanes 16-31,
    // column = 16*t + l16. Spill to global h and to LDS for the alpha reduction.
    #pragma unroll
    for (int t = 0; t < 4; ++t) {
        #pragma unroll
        for (int r = 0; r < 8; ++r) {
            int m   = r + 8 * hf;
            int row = rowbase + m;
            float v = acc[t][r];
            sH[wave][m][16 * t + l16] = v;
            if (row < n) h[(size_t)row * DIM + 16 * t + l16] = v;
        }
    }
    // wave-local LDS write->read; DScnt waits inserted by compiler

    // lanes 0-15 compute alpha_src for rows 0..15, lanes 16-31 compute alpha_dst
    int row = rowbase + l16;
    const float* av = hf ? avd : avs;
    float s = 0.f;
    #pragma unroll 8
    for (int j = 0; j < DIM; ++j) s += sH[wave][l16][j] * av[j];
    if (row < n) {
        if (hf) alpha_d[row] = s;
        else    alpha_s[row] = s;
    }
}

// ---------------------------------------------------------------------------
// Kernel 2: zero the per-node reduction buffers and the output accumulator
// ---------------------------------------------------------------------------
__global__ void gat_init(unsigned* __restrict__ maxu, float* __restrict__ denom,
                         float* __restrict__ outb, int n)
{
    int i = blockIdx.x * blockDim.x + threadIdx.x;
    if (i < n) { maxu[i] = 0u; denom[i] = 0.f; }
    if (i < n * DIM) outb[i] = 0.f;
}

// ---------------------------------------------------------------------------
// Kernel 3: segment max of leaky_relu(alpha_s[src] + alpha_d[dst]) over dst
// ---------------------------------------------------------------------------
__global__ void gat_edge_max(const long long* __restrict__ pos, const long long* __restrict__ neg,
                             long long ne, int n,
                             const float* __restrict__ as_, const float* __restrict__ ad_,
                             unsigned* __restrict__ maxu)
{
    long long e = (long long)blockIdx.x * blockDim.x + threadIdx.x;
    if (e >= 2 * ne + n) return;
    int s, d; get_edge(e, pos, neg, ne, s, d);
    float ev = as_[s] + ad_[d];
    ev = ev > 0.f ? ev : NEG_SLOPE * ev;
    atomicMax(&maxu[d], mapf(ev));
}

// ---------------------------------------------------------------------------
// Kernel 4: segment sum of exp(e - e_max[dst]) over dst
// ---------------------------------------------------------------------------
__global__ void gat_edge_sum(const long long* __restrict__ pos, const long long* __restrict__ neg,
                             long long ne, int n,
                             const float* __restrict__ as_, const float* __restrict__ ad_,
                             const unsigned* __restrict__ maxu, float* __restrict__ denom)
{
    long long e = (long long)blockIdx.x * blockDim.x + threadIdx.x;
    if (e >= 2 * ne + n) return;
    int s, d; get_edge(e, pos, neg, ne, s, d);
    float ev = as_[s] + ad_[d];
    ev = ev > 0.f ? ev : NEG_SLOPE * ev;
    float ex = __expf(ev - unmapf(maxu[d]));
    atomicAdd(&denom[d], ex);
}

// ---------------------------------------------------------------------------
// Kernel 5: out[dst] += h[src] * alpha  (one wave per edge, 2 dims per lane)
// ---------------------------------------------------------------------------
__global__ __launch_bounds__(256) void gat_edge_scatter(
    const long long* __restrict__ pos, const long long* __restrict__ neg,
    long long ne, int n,
    const float* __restrict__ as_, const float* __restrict__ ad_,
    const unsigned* __restrict__ maxu, const float* __restrict__ denom,
    const float* __restrict__ h, float* __restrict__ outb)
{
    int wave = threadIdx.x >> 5;
    int lane = threadIdx.x & 31;
    long long e = (long long)blockIdx.x * WAVES_PER_BLOCK + wave;
    if (e >= 2 * ne + n) return;   // wave-uniform
    int s, d; get_edge(e, pos, neg, ne, s, d);
    float ev = as_[s] + ad_[d];
    ev = ev > 0.f ? ev : NEG_SLOPE * ev;
    float ex = __expf(ev - unmapf(maxu[d]));
    float al = ex / (denom[d] + 1e-16f);
    const float* hs = h + (size_t)s * DIM;
    float*       od = outb + (size_t)d * DIM;
    atomicAdd(&od[lane],      hs[lane]      * al);
    atomicAdd(&od[lane + 32], hs[lane + 32] * al);
}

// ---------------------------------------------------------------------------
// Kernel 6: add per-layer bias
// ---------------------------------------------------------------------------
__global__ void gat_bias(float* __restrict__ outb, const float* __restrict__ b, int n)
{
    int i = blockIdx.x * blockDim.x + threadIdx.x;
    if (i < n * DIM) outb[i] += b[i & (DIM - 1)];
}

// ---------------------------------------------------------------------------
// Host-side launcher
// ---------------------------------------------------------------------------
extern "C" void kernel_launch(void* const* d_in, const int* in_sizes, int n_in,
                              void* d_out, int out_size, void* d_ws, size_t ws_size,
                              hipStream_t stream)
{
    const float*     x    = (const float*)d_in[0];       // [N, 64]
    const float*     Wl   = (const float*)d_in[1];       // [L, 64, 64]
    const float*     asl  = (const float*)d_in[2];       // [L, 64]
    const float*     adl  = (const float*)d_in[3];       // [L, 64]
    const float*     bl   = (const float*)d_in[4];       // [L, 64]
    const long long* pos  = (const long long*)d_in[5];   // [2, E] int64
    const long long* neg  = (const long long*)d_in[6];   // [2, E] int64

    int       n       = in_sizes[0] / DIM;
    long long ne      = (long long)in_sizes[5] / 2;
    int       nlayers = in_sizes[1] / (DIM * DIM);
    long long etot    = 2 * ne + (long long)n;

    // workspace carve-up
    char* wsp = (char*)d_ws;
    float*    hbuf  = (float*)wsp;  wsp += (size_t)n * DIM * sizeof(float);
    float*    tmp   = (float*)wsp;  wsp += (size_t)n * DIM * sizeof(float);
    float*    as_   = (float*)wsp;  wsp += (size_t)n * sizeof(float);
    float*    ad_   = (float*)wsp;  wsp += (size_t)n * sizeof(float);
    unsigned* maxu  = (unsigned*)wsp; wsp += (size_t)n * sizeof(unsigned);
    float*    denom = (float*)wsp;  wsp += (size_t)n * sizeof(float);

    dim3 blk(256);
    unsigned gGemm = (unsigned)((n + ROWS_PER_BLOCK - 1) / ROWS_PER_BLOCK);
    unsigned gNode = (unsigned)((n * DIM + 255) / 256);
    unsigned gEdge = (unsigned)((etot + 255) / 256);
    unsigned gScat = (unsigned)((etot + WAVES_PER_BLOCK - 1) / WAVES_PER_BLOCK);

    const float* zin = x;
    for (int l = 0; l < nlayers; ++l) {
        float* zout = (l == nlayers - 1) ? (float*)d_out : tmp;
        gat_gemm_alpha<<<gGemm, blk, 0, stream>>>(zin, Wl + (size_t)l * DIM * DIM,
                                                  asl + l * DIM, adl + l * DIM,
                                                  hbuf, as_, ad_, n);
        gat_init<<<gNode, blk, 0, stream>>>(maxu, denom, zout, n);
        gat_edge_max<<<gEdge, blk, 0, stream>>>(pos, neg, ne, n, as_, ad_, maxu);
        gat_edge_sum<<<gEdge, blk, 0, stream>>>(pos, neg, ne, n, as_, ad_, maxu, denom);
        gat_edge_scatter<<<gScat, blk, 0, stream>>>(pos, neg, ne, n, as_, ad_, maxu, denom,
                                                    hbuf, zout);
        gat_bias<<<gNode, blk, 0, stream>>>(zout, bl + l * DIM, n);
        zin = zout;
    }
}